// Func_Wrapper_22531398435124
// MI455X (gfx1250) — compile-verified
//
#include <hip/hip_runtime.h>

typedef __attribute__((ext_vector_type(2))) float v2f;
typedef __attribute__((ext_vector_type(8))) float v8f;

static __device__ __forceinline__ v8f wmma_f32(v2f a, v2f b, v8f c) {
    // D = A(16x4,f32) * B(4x16,f32) + C(16x16,f32)
    return __builtin_amdgcn_wmma_f32_16x16x4_f32(false, a, false, b,
                                                 (short)0, c, false, false);
}

static __device__ __forceinline__ float fast_tanh(float x) {
#if __has_builtin(__builtin_amdgcn_tanhf)
    return __builtin_amdgcn_tanhf(x);   // v_tanh_f32: single TRANS op, no branches
#else
    return tanhf(x);
#endif
}

// One wave32 handles a 16-row batch tile end-to-end:
//   GEMM1: u = z @ W1[1:,:]  (C init = t*W1[0,:] + b1), h = tanh(u)
//   trace: s[b] = sum_k (1-h^2)*c_k,  c_k = sum_d W1[1+d,k]*W2[k,d]
//   GEMM2: dz = h @ W2 + b2
//   f = -t*(z - scale*dz),  dlogp = t*(D - scale*s)
__global__ __launch_bounds__(32) void ffjord_exact_trace_kernel(
    const float* __restrict__ t_in, const float* __restrict__ z,
    const float* __restrict__ W1, const float* __restrict__ b1,
    const float* __restrict__ W2, const float* __restrict__ b2,
    float* __restrict__ f_out, float* __restrict__ dlogp_out)
{
    constexpr int D = 32, H = 256;
    __shared__ float sc[H];        // c_k per hidden unit
    __shared__ float sh[16 * H];   // h tile: 16 rows x 256 cols, fp32

    const int lane  = threadIdx.x;      // 0..31
    const int m     = lane & 15;        // matrix row (A) / column (B,C,D)
    const int khalf = lane >> 4;        // which half of the K=4 / M=16 split
    const int row0  = blockIdx.x * 16;  // batch tile origin

    // ---- precompute c_k = sum_d W1[1+d,k] * W2[k,d]  (8 k's per lane) ----
    #pragma unroll
    for (int i = 0; i < 8; ++i) {
        const int n = lane * 8 + i;
        float acc = 0.0f;
        #pragma unroll
        for (int d = 0; d < D; ++d)
            acc = fmaf(W1[(1 + d) * H + n], W2[n * D + d], acc);
        sc[n] = acc;
    }

    const float t     = t_in[0];
    const float scale = 1.0f / sqrtf(1.0f - expf(-t * t));

    // ---- A fragments for GEMM1: 8 K-chunks of the 16x32 z tile ----
    // A(16x4) lane layout: lanes 0-15 -> K={0,1}, lanes 16-31 -> K={2,3}
    const float* zrow = z + (size_t)(row0 + m) * D;
    v2f a1[8];
    #pragma unroll
    for (int kc = 0; kc < 8; ++kc) {
        const int k = 4 * kc + 2 * khalf;
        a1[kc].x = zrow[k];
        a1[kc].y = zrow[k + 1];
    }

    __syncthreads();  // sc visible

    float sacc[8] = {0, 0, 0, 0, 0, 0, 0, 0};  // trace partials per C/D VGPR

    // ---- GEMM1 over 16 N-tiles (H=256), K=32 in 8 chunks of 4 ----
    for (int nt = 0; nt < 16; ++nt) {
        const int   n  = nt * 16 + m;                 // hidden column this lane owns
        const float c0 = fmaf(t, W1[n], b1[n]);       // t*W1[0,n] + b1[n]
        v8f acc = {c0, c0, c0, c0, c0, c0, c0, c0};
        #pragma unroll
        for (int kc = 0; kc < 8; ++kc) {
            const int kk = 4 * kc + 2 * khalf;
            v2f bf;
            bf.x = W1[(1 + kk) * H + n];              // B[q,n] = W1[1+k, n]
            bf.y = W1[(2 + kk) * H + n];
            acc = wmma_f32(a1[kc], bf, acc);
        }
        // tanh, trace partial, stage h into LDS (row = v + 8*khalf, col = n)
        const float cn = sc[n];
        #pragma unroll
        for (int v = 0; v < 8; ++v) {
            const float hv = fast_tanh(acc[v]);
            sacc[v] = fmaf(1.0f - hv * hv, cn, sacc[v]);
            sh[(v + 8 * khalf) * H + n] = hv;
        }
    }

    __syncthreads();  // h tile visible across lanes

    // ---- dlogp: reduce trace partials within each 16-lane half ----
    #pragma unroll
    for (int v = 0; v < 8; ++v) {
        float s = sacc[v];
        s += __shfl_xor(s, 1, 32);
        s += __shfl_xor(s, 2, 32);
        s += __shfl_xor(s, 4, 32);
        s += __shfl_xor(s, 8, 32);
        if (m == 0)
            dlogp_out[row0 + v + 8 * khalf] = t * ((float)D - scale * s);
    }

    // ---- GEMM2: dz = h @ W2 + b2 ; M=16, N=32 (2 tiles), K=256 (64 chunks) ----
    #pragma unroll 2
    for (int nt2 = 0; nt2 < 2; ++nt2) {
        const int   n2 = nt2 * 16 + m;                // output column 0..31
        const float c0 = b2[n2];
        v8f acc = {c0, c0, c0, c0, c0, c0, c0, c0};
        const float* hrow = &sh[m * H];               // A row for this lane
        #pragma unroll 8
        for (int kc = 0; kc < 64; ++kc) {
            const int kk = 4 * kc + 2 * khalf;
            v2f af, bf;
            af.x = hrow[kk];                          // A[m, k] = h[m, k]
            af.y = hrow[kk + 1];
            bf.x = W2[kk * D + n2];                   // B[k, n] = W2[k, n]
            bf.y = W2[(kk + 1) * D + n2];
            acc = wmma_f32(af, bf, acc);
        }
        // f = -t * (z - scale * dz)
        #pragma unroll
        for (int v = 0; v < 8; ++v) {
            const int    row = row0 + v + 8 * khalf;
            const size_t gi  = (size_t)row * D + n2;
            f_out[gi] = -t * (z[gi] - scale * acc[v]);
        }
    }
}

extern "C" void kernel_launch(void* const* d_in, const int* in_sizes, int n_in,
                              void* d_out, int out_size, void* d_ws, size_t ws_size,
                              hipStream_t stream) {
    const float* t  = (const float*)d_in[0];
    const float* z  = (const float*)d_in[1];
    const float* W1 = (const float*)d_in[2];
    const float* b1 = (const float*)d_in[3];
    const float* W2 = (const float*)d_in[4];
    const float* b2 = (const float*)d_in[5];

    const int B = in_sizes[1] / 32;   // 4096
    float* f_out = (float*)d_out;
    float* dlogp = f_out + (size_t)B * 32;

    ffjord_exact_trace_kernel<<<B / 16, 32, 0, stream>>>(t, z, W1, b1, W2, b2,
                                                         f_out, dlogp);
    (void)n_in; (void)d_ws; (void)ws_size; (void)out_size;
}